// AutoCorrelation_35665408426441
// MI455X (gfx1250) — compile-verified
//
#include <hip/hip_runtime.h>
#include <math.h>

#define L_LEN   4096
#define D_DIM   64
#define NHEADS  64      // B*H = 4*16
#define KMAX    64
#define NTILES  (L_LEN / 16)     // 256 K-tiles per head
#define TSTRIDE 68               // staged-tile row stride (floats): bank-conflict-free, 8B aligned

typedef __attribute__((ext_vector_type(2))) float v2f;
typedef __attribute__((ext_vector_type(4))) float v4f;
typedef __attribute__((ext_vector_type(8))) float v8f;
typedef __attribute__((ext_vector_type(2))) int   v2i;

// Pointer types matching the async-DMA builtin prototype: (v2i AS1*, v2i AS3*, imm, imm)
typedef v2i __attribute__((address_space(1)))* gbl64_t;
typedef v2i __attribute__((address_space(3)))* lds64_t;

// ---------------------------------------------------------------------------
// Async staging of one 16x64 f32 K-tile (4KB) into LDS, row stride TSTRIDE.
// 256 threads x 2 x 8B copies. Uses gfx1250 GLOBAL_LOAD_ASYNC_TO_LDS_B64
// (ASYNCcnt-tracked DMA) when the builtin exists; falls back to plain copies.
// ---------------------------------------------------------------------------
__device__ __forceinline__ void stage_tile(const float* __restrict__ khead,
                                           float* lds_buf, int t0, int tid) {
#pragma unroll
    for (int rep = 0; rep < 2; ++rep) {
        const int u   = tid + rep * 256;          // 0..511 : 8B chunk index
        const int row = u >> 5;                   // 0..15
        const int c   = u & 31;                   // 0..31  (8B chunks in a row)
        float* g = (float*)(khead + (long)(t0 + row) * D_DIM + c * 2);
        float* l = lds_buf + row * TSTRIDE + c * 2;
#if __has_builtin(__builtin_amdgcn_global_load_async_to_lds_b64)
        __builtin_amdgcn_global_load_async_to_lds_b64((gbl64_t)g, (lds64_t)l, 0, 0);
#else
        *(v2f*)l = *(const v2f*)g;
#endif
    }
}

__device__ __forceinline__ void wait_async_done() {
#if __has_builtin(__builtin_amdgcn_s_wait_asynccnt)
    __builtin_amdgcn_s_wait_asynccnt(0);
#else
    __asm__ volatile("s_wait_asynccnt 0" ::: "memory");
#endif
}

// ---------------------------------------------------------------------------
// Kernel 1: corr[head][tau] = sum_s <Q[s,:], K[(s-tau)%L,:]>   (un-normalized)
// Q*K^T in 16x16 f32 WMMA tiles; tile entry (m,n) -> diagonal
// tau = (s0+m - t0-n) mod L, accumulated in LDS, flushed atomically.
// B-operand (K tile) is DMA-staged to LDS once per block, double-buffered.
// ---------------------------------------------------------------------------
__global__ __launch_bounds__(256)
void corr_wmma_kernel(const float* __restrict__ Q,
                      const float* __restrict__ Kp,
                      float* __restrict__ corr) {
    __shared__ float s_corr[L_LEN];               // 16 KB diagonal accumulator
    __shared__ float s_kt[2][16 * TSTRIDE];       // 2 x 4.25 KB staged K tiles

    const int head  = blockIdx.x;                 // 0..63
    const int chunk = blockIdx.y;                 // 0..31 (128 Q rows each)
    const int tid   = threadIdx.x;
    const int wave  = tid >> 5;                   // 8 waves = 8 m-tiles
    const int lane  = tid & 31;
    const int half  = lane >> 4;                  // lane group (K-dim split)
    const int i     = lane & 15;                  // row-in-tile / N index

    const long base = (long)head * L_LEN * D_DIM;
    const float* khead = Kp + base;

    stage_tile(khead, s_kt[0], 0, tid);           // prefetch tile 0 (async DMA)

    for (int t = tid; t < L_LEN; t += 256) s_corr[t] = 0.0f;

    // Preload A (this wave's Q tile): a[kk] = k-dims {4kk+2*half, +1} of row s0+i
    const int s0 = chunk * 128 + wave * 16;
    v2f a[16];
    {
        const float* qrow = Q + base + (long)(s0 + i) * D_DIM + 2 * half;
#pragma unroll
        for (int kk = 0; kk < 16; ++kk)
            a[kk] = *(const v2f*)(qrow + 4 * kk);
    }

    for (int nt = 0; nt < NTILES; ++nt) {
        const int cur = nt & 1;
        wait_async_done();                        // my stage of tile nt complete
        __syncthreads();                          // all waves staged; prev reads done
        if (nt + 1 < NTILES)                      // overlap DMA of nt+1 with compute
            stage_tile(khead, s_kt[cur ^ 1], (nt + 1) * 16, tid);

        const int    t0   = nt * 16;
        const float* brow = &s_kt[cur][i * TSTRIDE + 2 * half];

        // Hoist all B operands: 8x ds_load_2addr_b64 issued as one clause so
        // LDS latency overlaps the WMMA stream (partial dscnt waits).
        v2f b[16];
#pragma unroll
        for (int kk = 0; kk < 16; ++kk)
            b[kk] = *(const v2f*)(brow + 4 * kk);

        // Two independent accumulator chains -> back-to-back WMMA issue slack.
        v8f c0 = {}, c1 = {};
#pragma unroll
        for (int kk = 0; kk < 16; kk += 2) {
            c0 = __builtin_amdgcn_wmma_f32_16x16x4_f32(
                     false, a[kk],     false, b[kk],     (short)0, c0, false, false);
            c1 = __builtin_amdgcn_wmma_f32_16x16x4_f32(
                     false, a[kk + 1], false, b[kk + 1], (short)0, c1, false, false);
        }
        const v8f c = c0 + c1;

        // Diagonal scatter: VGPR r -> (m = r + 8*half, n = i); <=2-way collisions.
#pragma unroll
        for (int r = 0; r < 8; ++r) {
            const int m   = r + 8 * half;
            const int tau = (s0 + m - t0 - i) & (L_LEN - 1);
            atomicAdd(&s_corr[tau], c[r]);
        }
    }
    __syncthreads();

    float* dst = corr + (long)head * L_LEN;
    for (int t = tid; t < L_LEN; t += 256)
        atomicAdd(&dst[t], s_corr[t]);
}

// ---------------------------------------------------------------------------
// Kernel 2: per-head top-k over L (tie -> lowest index, like jax top_k),
// then softmax over the k peak values. k = min(int(factor*ln(L)), L, KMAX).
// ---------------------------------------------------------------------------
__global__ __launch_bounds__(256)
void topk_softmax_kernel(const float* __restrict__ corr,
                         const int* __restrict__ factor,
                         int* __restrict__ delays,
                         float* __restrict__ weights) {
    __shared__ float sc[L_LEN];
    __shared__ float rv[256];
    __shared__ int   ri[256];
    __shared__ float wloc[KMAX];
    __shared__ int   dloc[KMAX];

    const int head = blockIdx.x;
    const int tid  = threadIdx.x;

    int k = (int)((float)factor[0] * logf((float)L_LEN));
    if (k > KMAX)  k = KMAX;
    if (k > L_LEN) k = L_LEN;

    const float* src = corr + (long)head * L_LEN;
    for (int t = tid; t < L_LEN; t += 256)
        sc[t] = src[t] * (1.0f / (float)D_DIM);   // mean over D
    __syncthreads();

    for (int round = 0; round < k; ++round) {
        float bv = -INFINITY; int bi = 0x7fffffff;
        for (int t = tid; t < L_LEN; t += 256) {
            float v = sc[t];
            if (v > bv || (v == bv && t < bi)) { bv = v; bi = t; }
        }
        rv[tid] = bv; ri[tid] = bi;
        __syncthreads();
        for (int off = 128; off > 0; off >>= 1) {
            if (tid < off) {
                float ov = rv[tid + off]; int oi = ri[tid + off];
                if (ov > rv[tid] || (ov == rv[tid] && oi < ri[tid])) {
                    rv[tid] = ov; ri[tid] = oi;
                }
            }
            __syncthreads();
        }
        if (tid == 0) {
            dloc[round] = ri[0];
            wloc[round] = rv[0];
            sc[ri[0]]   = -INFINITY;              // mask chosen peak
        }
        __syncthreads();
    }

    if (tid == 0) {
        float mx = wloc[0];
        for (int j = 1; j < k; ++j) mx = fmaxf(mx, wloc[j]);
        float s = 0.0f;
        for (int j = 0; j < k; ++j) { wloc[j] = expf(wloc[j] - mx); s += wloc[j]; }
        const float inv = 1.0f / s;
        for (int j = 0; j < k; ++j) {
            weights[head * KMAX + j] = wloc[j] * inv;
            delays [head * KMAX + j] = dloc[j];
        }
    }
}

// ---------------------------------------------------------------------------
// Kernel 3: out[t,d] = sum_j w[j] * V[(t - delay_j) mod L, d]   (per head)
// Memory-bound gather (~1 GB reads, L2-resident): b128 loads/stores.
// ---------------------------------------------------------------------------
__global__ __launch_bounds__(256)
void rolled_sum_kernel(const float* __restrict__ V,
                       const int* __restrict__ factor,
                       const int* __restrict__ delays,
                       const float* __restrict__ weights,
                       float* __restrict__ out) {
    __shared__ float w_s[KMAX];
    __shared__ int   d_s[KMAX];

    const int head = blockIdx.x;
    const int tblk = blockIdx.y;                  // 32 blocks of 128 t
    const int tid  = threadIdx.x;

    int k = (int)((float)factor[0] * logf((float)L_LEN));
    if (k > KMAX)  k = KMAX;
    if (k > L_LEN) k = L_LEN;

    if (tid < k) {
        w_s[tid] = weights[head * KMAX + tid];
        d_s[tid] = delays [head * KMAX + tid];
    }
    __syncthreads();

    const long base = (long)head * L_LEN * D_DIM;
    const int  d4   = (tid & 15) * 4;             // float4 column
    const int  trow = tid >> 4;                   // 0..15

    for (int t = tblk * 128 + trow; t < tblk * 128 + 128; t += 16) {
        v4f acc = {};
        for (int j = 0; j < k; ++j) {
            const int src_t = (t - d_s[j]) & (L_LEN - 1);
            const v4f v = *(const v4f*)&V[base + (long)src_t * D_DIM + d4];
            acc += w_s[j] * v;                    // global_load_b128 + v_fma
        }
        *(v4f*)&out[base + (long)t * D_DIM + d4] = acc;  // global_store_b128
    }
}

// ---------------------------------------------------------------------------
extern "C" void kernel_launch(void* const* d_in, const int* in_sizes, int n_in,
                              void* d_out, int out_size, void* d_ws, size_t ws_size,
                              hipStream_t stream) {
    const float* Q      = (const float*)d_in[0];
    const float* Kp     = (const float*)d_in[1];
    const float* V      = (const float*)d_in[2];
    const int*   factor = (const int*)d_in[3];
    float*       out    = (float*)d_out;

    // Workspace layout: corr (1 MB) | delays | weights
    float* corr = (float*)d_ws;
    const size_t corr_bytes = (size_t)NHEADS * L_LEN * sizeof(float);
    int*   delays  = (int*)((char*)d_ws + corr_bytes);
    float* weights = (float*)((char*)d_ws + corr_bytes + (size_t)NHEADS * KMAX * sizeof(int));

    (void)hipMemsetAsync(corr, 0, corr_bytes, stream);  // graph-capturable

    corr_wmma_kernel   <<<dim3(NHEADS, 32), 256, 0, stream>>>(Q, Kp, corr);
    topk_softmax_kernel<<<dim3(NHEADS),     256, 0, stream>>>(corr, factor, delays, weights);
    rolled_sum_kernel  <<<dim3(NHEADS, 32), 256, 0, stream>>>(V, factor, delays, weights, out);
}